// MyConv2d1_49718541419170
// MI455X (gfx1250) — compile-verified
//
#include <hip/hip_runtime.h>

typedef __attribute__((ext_vector_type(2))) float v2f;
typedef __attribute__((ext_vector_type(8))) float v8f;

#define B_    32
#define CIN   128
#define HH    56
#define WW    56
#define COUT  256
#define KTOT  1152                 // CIN * 9
#define HW    3136                 // 56*56
#define NPIX  (B_ * HW)            // 100352

// One workgroup = 256 threads = 8 wave32.
// Block computes a [128 co] x [32 pixel] output tile via implicit GEMM.
// Each wave: one 16-co row, TWO 16-pixel accumulators sharing each A load.
// LDS im2col layout [k/4][tile][n][k%4]: the two B operands for one k-step
// are 256 B apart -> single ds_load_2addr_b64 per k-step per lane.
// K = 1152 ordered as k = tap*128 + ci  (tap = kh*3+kw).
__global__ __launch_bounds__(256) void conv3x3_wmma_f32(
    const float* __restrict__ x,     // [B, CIN, H, W]
    const float* __restrict__ wgt,   // [COUT, CIN, 3, 3]
    const float* __restrict__ bias,  // [COUT]
    float* __restrict__ out)         // [B, COUT, H, W]
{
    __shared__ float patch[2 * KTOT * 16];   // 147456 bytes

    const int t       = threadIdx.x;
    const int n_base  = blockIdx.x * 32;
    const int co_blk  = blockIdx.y * 128;

    // ---------------- Phase 1: gather im2col patches into LDS ----------------
    {
        const int p    = t & 31;                 // pixel 0..31 owned by this thread
        const int n    = n_base + p;
        const int b    = n / HW;
        const int rem  = n % HW;
        const int h    = rem / WW;
        const int wpx  = rem % WW;
        const int ksl  = t >> 5;                 // 0..7, k-slice owned by this thread
        const float* __restrict__ xb = x + (size_t)b * CIN * HW;
        // [k/4][tile][n][k%4]
        const int base = ((p >> 4) << 6) + ((p & 15) << 2);

        #pragma unroll 4
        for (int iter = 0; iter < KTOT / 8; ++iter) {   // 144 iterations
            const int k   = ksl + (iter << 3);          // 0..1151
            const int tap = k >> 7;                     // 0..8
            const int ci  = k & 127;
            const int kh  = tap / 3;
            const int kw  = tap - kh * 3;
            const int ih  = h + kh - 1;
            const int iw  = wpx + kw - 1;
            float v = 0.0f;
            if ((unsigned)ih < (unsigned)HH && (unsigned)iw < (unsigned)WW)
                v = xb[ci * HW + ih * WW + iw];
            patch[((k >> 2) << 7) + base + (k & 3)] = v;
        }
    }
    __syncthreads();

    // ---------------- Phase 2: per-wave 16x32 WMMA GEMM over K=1152 ----------
    const int lane   = t & 31;
    const int wv     = t >> 5;                  // 0..7
    const int co_t   = co_blk + wv * 16;
    const int m      = lane & 15;               // A row / B column / D column
    const int halfv  = lane >> 4;               // lane half selects K pair

    // A 16x4 fp32 layout: lanes 0-15 -> {K0,K1} in regs {0,1}; lanes 16-31 -> {K2,K3}
    const float* __restrict__ wrow = wgt + (size_t)(co_t + m) * KTOT;
    const v2f*   __restrict__ bp   = (const v2f*)patch;
    const int    bm_lane = (m << 1) + halfv;    // per-lane B offset within a chunk

    v8f acc0 = {};
    v8f acc1 = {};
    for (int tap = 0; tap < 9; ++tap) {
        const int wk0   = tap + 18 * halfv;     // (2*halfv)*9 + tap
        const int kbase = tap << 7;
        #pragma unroll 4
        for (int cb = 0; cb < 128; cb += 4) {
            v2f a, bm0, bm1;
            const int wi = wk0 + cb * 9;        // ci stride is 9 in OIHW flat
            a.x = wrow[wi];
            a.y = wrow[wi + 9];
            // B pair K = 2*halfv + {0,1}, pixel column m; tiles 256 B apart
            const int bofs = (((kbase + cb) >> 2) << 6) + bm_lane;
            bm0 = bp[bofs];
            bm1 = bp[bofs + 32];
            // (neg_a, A, neg_b, B, c_mod, C, reuse_a, reuse_b)
            acc0 = __builtin_amdgcn_wmma_f32_16x16x4_f32(
                false, a, false, bm0, (short)0, acc0, false, false);
            acc1 = __builtin_amdgcn_wmma_f32_16x16x4_f32(
                false, a, false, bm1, (short)0, acc1, false, false);
        }
    }

    // ---------------- Epilogue: bias add + NCHW scatter ----------------------
    // D layout (32-bit 16x16): VGPR r -> M = r + 8*lane_half, N = lane&15.
    const int n0    = n_base + m;
    const int n1    = n_base + 16 + m;
    const int b0    = n0 / HW;
    const int b1    = n1 / HW;
    const size_t pix0 = (size_t)b0 * COUT * HW + (n0 % HW);
    const size_t pix1 = (size_t)b1 * COUT * HW + (n1 % HW);
    #pragma unroll
    for (int r = 0; r < 8; ++r) {
        const int co = co_t + r + 8 * halfv;
        const float bb = bias[co];
        out[pix0 + (size_t)co * HW] = acc0[r] + bb;
        out[pix1 + (size_t)co * HW] = acc1[r] + bb;
    }
}

extern "C" void kernel_launch(void* const* d_in, const int* in_sizes, int n_in,
                              void* d_out, int out_size, void* d_ws, size_t ws_size,
                              hipStream_t stream) {
    const float* x    = (const float*)d_in[0];
    const float* w    = (const float*)d_in[1];
    const float* bias = (const float*)d_in[2];
    float* out        = (float*)d_out;

    dim3 grid(NPIX / 32, COUT / 128);   // 3136 x 2 blocks
    conv3x3_wmma_f32<<<grid, 256, 0, stream>>>(x, w, bias, out);
}